// PointNet2_1168231104859
// MI455X (gfx1250) — compile-verified
//
#include <hip/hip_runtime.h>
#include <hip/hip_bf16.h>

typedef float v2f __attribute__((ext_vector_type(2)));
typedef float v8f __attribute__((ext_vector_type(8)));

#define EPS_BN 1e-5f

// ---------------------------------------------------------------------------
// GEMM + folded BN/ReLU epilogue.  Y[M,O] = act( X[M,C] * W[O,C]^T * s + bias )
// One wave per 16x16 output tile, K stepped by 4 via V_WMMA_F32_16X16X4_F32.
// store_t==1: store transposed per batch: Y[((m/Npts)*O + n)*Npts + m%Npts]
// act==1: folded BatchNorm + ReLU; act==0: bias only.
// ---------------------------------------------------------------------------
__global__ void gemm_mlp_kernel(const float* __restrict__ X, int M, int C,
                                const float* __restrict__ Wm,
                                const float* __restrict__ cb,
                                const float* __restrict__ gg,
                                const float* __restrict__ bt,
                                const float* __restrict__ mu,
                                const float* __restrict__ var,
                                int O, float* __restrict__ Y,
                                int act, int store_t, int Npts)
{
    int tilesN = (O + 15) >> 4;
    int tilesM = (M + 15) >> 4;
    long tiles = (long)tilesM * tilesN;
    long wave  = (long)blockIdx.x * (blockDim.x >> 5) + (threadIdx.x >> 5);
    if (wave >= tiles) return;                 // wave-uniform: EXEC stays full
    int lane = threadIdx.x & 31;
    int tm = (int)(wave / tilesN);
    int tn = (int)(wave % tilesN);
    int m0 = tm << 4, n0 = tn << 4;
    int l16 = lane & 15;
    int hi  = lane >> 4;                       // 0 or 1
    int kb  = hi << 1;                         // 0 or 2

    int rowL = m0 + l16; if (rowL >= M) rowL = M - 1;
    int colL = n0 + l16; if (colL >= O) colL = O - 1;
    const float* xr = X  + (size_t)rowL * C;
    const float* wr = Wm + (size_t)colL * C;

    v8f acc = {0.f,0.f,0.f,0.f,0.f,0.f,0.f,0.f};
    int k = 0;
    for (; k + 4 <= C; k += 4) {
        v2f a, b;
        a.x = xr[k + kb];  a.y = xr[k + kb + 1];
        b.x = wr[k + kb];  b.y = wr[k + kb + 1];
        acc = __builtin_amdgcn_wmma_f32_16x16x4_f32(false, a, false, b,
                                                    (short)0, acc, false, false);
    }
    if (k < C) {                               // K tail, zero-padded
        v2f a = {0.f, 0.f}, b = {0.f, 0.f};
        if (k + kb     < C) { a.x = xr[k + kb];     b.x = wr[k + kb];     }
        if (k + kb + 1 < C) { a.y = xr[k + kb + 1]; b.y = wr[k + kb + 1]; }
        acc = __builtin_amdgcn_wmma_f32_16x16x4_f32(false, a, false, b,
                                                    (short)0, acc, false, false);
    }

    float s = 1.f, bias;
    if (act) {
        float sc = gg[colL] * rsqrtf(var[colL] + EPS_BN);
        s = sc;
        bias = (cb[colL] - mu[colL]) * sc + bt[colL];
    } else {
        bias = cb[colL];
    }

    int n = n0 + l16;
#pragma unroll
    for (int r = 0; r < 8; ++r) {
        int m = m0 + r + (hi << 3);
        if (m < M && n < O) {
            float y = acc[r] * s + bias;
            if (act) y = fmaxf(y, 0.f);
            if (store_t) {
                int b_ = m / Npts, i_ = m % Npts;
                Y[((size_t)b_ * O + n) * Npts + i_] = y;
            } else {
                Y[(size_t)m * O + n] = y;
            }
        }
    }
}

// ---------------------------------------------------------------------------
// Farthest point sampling: 1 block per batch, replicates jax.lax.scan order
// (emit current `far`, update min-dist, argmax with lowest-index tie-break).
// ---------------------------------------------------------------------------
__global__ void fps_kernel(const float* __restrict__ xyz, int N, int npoint,
                           int* __restrict__ out)
{
    __shared__ float dist[4096];
    __shared__ float rv[256];
    __shared__ int   ri[256];
    int b   = blockIdx.x;
    int tid = threadIdx.x;
    const float* p = xyz + (size_t)b * N * 3;
    for (int i = tid; i < N; i += blockDim.x) dist[i] = 1e10f;
    __syncthreads();
    int far = 0;
    for (int it = 0; it < npoint; ++it) {
        if (tid == 0) out[b * npoint + it] = far;
        float cx = p[3*far], cy = p[3*far+1], cz = p[3*far+2];
        float best = -1.f; int bi = N;
        for (int i = tid; i < N; i += blockDim.x) {
            float dx = p[3*i] - cx, dy = p[3*i+1] - cy, dz = p[3*i+2] - cz;
            float d  = dx*dx + dy*dy + dz*dz;
            float nd = fminf(dist[i], d);
            dist[i] = nd;
            if (nd > best) { best = nd; bi = i; }
        }
        rv[tid] = best; ri[tid] = bi;
        __syncthreads();
        for (int sft = blockDim.x >> 1; sft > 0; sft >>= 1) {
            if (tid < sft) {
                if (rv[tid+sft] > rv[tid] ||
                    (rv[tid+sft] == rv[tid] && ri[tid+sft] < ri[tid])) {
                    rv[tid] = rv[tid+sft]; ri[tid] = ri[tid+sft];
                }
            }
            __syncthreads();
        }
        far = ri[0];
        __syncthreads();
    }
}

__global__ void gather_xyz_kernel(const float* __restrict__ xyz,
                                  const int* __restrict__ fidx,
                                  int total, int N, int S,
                                  float* __restrict__ nxyz)
{
    int g = blockIdx.x * blockDim.x + threadIdx.x;
    if (g >= total) return;
    int b = g / S;
    int j = fidx[g];
    const float* p = xyz + ((size_t)b * N + j) * 3;
    nxyz[3*(size_t)g    ] = p[0];
    nxyz[3*(size_t)g + 1] = p[1];
    nxyz[3*(size_t)g + 2] = p[2];
}

// First 32 in-radius indices in ascending index order, padded with first.
__global__ void ball_query_kernel(const float* __restrict__ xyz,
                                  const float* __restrict__ nxyz,
                                  int total, int N, int S, float r2,
                                  int* __restrict__ idx)
{
    int g = blockIdx.x * blockDim.x + threadIdx.x;
    if (g >= total) return;
    int b = g / S;
    const float* p = xyz + (size_t)b * N * 3;
    float cx = nxyz[3*(size_t)g], cy = nxyz[3*(size_t)g+1], cz = nxyz[3*(size_t)g+2];
    int* o = idx + (size_t)g * 32;
    int cnt = 0;
    for (int j = 0; j < N && cnt < 32; ++j) {
        float dx = p[3*j] - cx, dy = p[3*j+1] - cy, dz = p[3*j+2] - cz;
        if (dx*dx + dy*dy + dz*dz <= r2) o[cnt++] = j;
    }
    int first = (cnt > 0) ? o[0] : 0;
    for (int k2 = cnt; k2 < 32; ++k2) o[k2] = first;
}

// Build grouped input rows: [recentered xyz(3) | gathered feats(Cf)]
__global__ void group_build_kernel(const float* __restrict__ xyz,
                                   const float* __restrict__ feats,
                                   const float* __restrict__ nxyz,
                                   const int* __restrict__ idx,
                                   long total, int N, int S, int K, int Cf,
                                   float* __restrict__ Xg)
{
    long g = (long)blockIdx.x * blockDim.x + threadIdx.x;
    if (g >= total) return;
    long bs = g / K;
    int  b  = (int)(bs / S);
    int  j  = idx[g];
    const float* pt  = xyz  + ((size_t)b * N + j) * 3;
    const float* ctr = nxyz + (size_t)bs * 3;
    int Cin = 3 + Cf;
    float* row = Xg + (size_t)g * Cin;
    row[0] = pt[0] - ctr[0];
    row[1] = pt[1] - ctr[1];
    row[2] = pt[2] - ctr[2];
    if (feats) {
        const float* f = feats + ((size_t)b * N + j) * Cf;
        for (int c = 0; c < Cf; ++c) row[3 + c] = f[c];
    }
}

__global__ void maxpool_kernel(const float* __restrict__ Y, long total,
                               int K, int C, float* __restrict__ out)
{
    long g = (long)blockIdx.x * blockDim.x + threadIdx.x;
    if (g >= total) return;
    long bs = g / C;
    int  c  = (int)(g % C);
    const float* y = Y + (size_t)bs * K * C + c;
    float m = y[0];
    for (int k = 1; k < K; ++k) m = fmaxf(m, y[(size_t)k * C]);
    out[g] = m;
}

// 3 nearest neighbors + normalized inverse-distance weights.
__global__ void knn3_kernel(const float* __restrict__ x1,
                            const float* __restrict__ x2,
                            long total, int n1, int n2,
                            int* __restrict__ idx3, float* __restrict__ w3)
{
    long g = (long)blockIdx.x * blockDim.x + threadIdx.x;
    if (g >= total) return;
    int b = (int)(g / n1);
    const float* q = x1 + (size_t)g * 3;
    const float* p = x2 + (size_t)b * n2 * 3;
    float d0 = 1e30f, d1 = 1e30f, d2 = 1e30f;
    int   i0 = 0, i1 = 0, i2 = 0;
    for (int j = 0; j < n2; ++j) {
        float dx = p[3*j] - q[0], dy = p[3*j+1] - q[1], dz = p[3*j+2] - q[2];
        float d  = dx*dx + dy*dy + dz*dz;
        if      (d < d0) { d2=d1; i2=i1; d1=d0; i1=i0; d0=d; i0=j; }
        else if (d < d1) { d2=d1; i2=i1; d1=d;  i1=j; }
        else if (d < d2) { d2=d;  i2=j; }
    }
    float w0 = 1.f/(d0+1e-8f), w1 = 1.f/(d1+1e-8f), w2 = 1.f/(d2+1e-8f);
    float ws = w0 + w1 + w2;
    w3[(size_t)g*3+0] = w0/ws; w3[(size_t)g*3+1] = w1/ws; w3[(size_t)g*3+2] = w2/ws;
    idx3[(size_t)g*3+0] = i0;  idx3[(size_t)g*3+1] = i1;  idx3[(size_t)g*3+2] = i2;
}

__global__ void interp_kernel(const float* __restrict__ p2,
                              const int* __restrict__ idx3,
                              const float* __restrict__ w3,
                              long total, int n1, int n2, int C2, int Ctot,
                              int coff, float* __restrict__ Xc)
{
    long g = (long)blockIdx.x * blockDim.x + threadIdx.x;
    if (g >= total) return;
    long bi = g / C2;
    int  c  = (int)(g % C2);
    int  b  = (int)(bi / n1);
    const float* pb = p2 + (size_t)b * n2 * C2;
    const int*   id = idx3 + (size_t)bi * 3;
    const float* w  = w3   + (size_t)bi * 3;
    float v = w[0]*pb[(size_t)id[0]*C2 + c]
            + w[1]*pb[(size_t)id[1]*C2 + c]
            + w[2]*pb[(size_t)id[2]*C2 + c];
    Xc[(size_t)bi * Ctot + coff + c] = v;
}

__global__ void copyfeat_kernel(const float* __restrict__ p1, long total,
                                int C1, int Ctot, float* __restrict__ Xc)
{
    long g = (long)blockIdx.x * blockDim.x + threadIdx.x;
    if (g >= total) return;
    long bi = g / C1;
    int  c  = (int)(g % C1);
    Xc[(size_t)bi * Ctot + c] = p1[g];
}

// ---------------------------------------------------------------------------
// Host-side orchestration
// ---------------------------------------------------------------------------
struct Lyr { const float *w, *b, *g, *bt, *mu, *var; };

static inline Lyr getL(void* const* din, int base) {
    Lyr l;
    l.w  = (const float*)din[base + 0];
    l.b  = (const float*)din[base + 1];
    l.g  = (const float*)din[base + 2];
    l.bt = (const float*)din[base + 3];
    l.mu = (const float*)din[base + 4];
    l.var= (const float*)din[base + 5];
    return l;
}

static void run_gemm(const float* X, long M, int C, const Lyr& L, int O,
                     float* Y, int act, int store_t, int Npts, hipStream_t s)
{
    long tiles  = ((M + 15) / 16) * (long)((O + 15) / 16);
    long blocks = (tiles + 3) / 4;                 // 4 waves (128 threads)/block
    gemm_mlp_kernel<<<dim3((unsigned)blocks), 128, 0, s>>>(
        X, (int)M, C, L.w, L.b, L.g, L.bt, L.mu, L.var, O, Y, act, store_t, Npts);
}

static void run_sa(const float* xyz_in, int Nin, const float* feats, int Cf,
                   int S, float radius, const Lyr* layers, int nl, const int* Os,
                   float* nxyz_out, float* pooled_out,
                   int* fpsb, int* ballb, float* bufA, float* bufB,
                   hipStream_t stream)
{
    fps_kernel<<<8, 256, 0, stream>>>(xyz_in, Nin, S, fpsb);
    int BS = 8 * S;
    gather_xyz_kernel<<<(BS + 255) / 256, 256, 0, stream>>>(
        xyz_in, fpsb, BS, Nin, S, nxyz_out);
    ball_query_kernel<<<(BS + 255) / 256, 256, 0, stream>>>(
        xyz_in, nxyz_out, BS, Nin, S, radius * radius, ballb);
    long M = (long)BS * 32;
    group_build_kernel<<<(unsigned)((M + 255) / 256), 256, 0, stream>>>(
        xyz_in, feats, nxyz_out, ballb, M, Nin, S, 32, Cf, bufA);
    const float* src = bufA;
    float* dst = bufB;
    int Cin = 3 + Cf;
    for (int l = 0; l < nl; ++l) {
        run_gemm(src, M, Cin, layers[l], Os[l], dst, 1, 0, 0, stream);
        Cin = Os[l];
        src = dst;
        dst = (dst == bufB) ? bufA : bufB;
    }
    long tot = (long)BS * Os[nl - 1];
    maxpool_kernel<<<(unsigned)((tot + 255) / 256), 256, 0, stream>>>(
        src, tot, 32, Os[nl - 1], pooled_out);
}

static void run_fp(const float* x1, int n1, const float* x2, int n2,
                   const float* p1, int C1, const float* p2, int C2,
                   const Lyr* layers, int nl, const int* Os, float* out,
                   int* knni, float* knnw, float* bufA, float* bufB,
                   hipStream_t stream)
{
    long tot1 = 8L * n1;
    knn3_kernel<<<(unsigned)((tot1 + 255) / 256), 256, 0, stream>>>(
        x1, x2, tot1, n1, n2, knni, knnw);
    int Ctot = C1 + C2;
    if (C1 > 0) {
        long tc = tot1 * C1;
        copyfeat_kernel<<<(unsigned)((tc + 255) / 256), 256, 0, stream>>>(
            p1, tc, C1, Ctot, bufA);
    }
    long ti = tot1 * C2;
    interp_kernel<<<(unsigned)((ti + 255) / 256), 256, 0, stream>>>(
        p2, knni, knnw, ti, n1, n2, C2, Ctot, C1, bufA);
    const float* src = bufA;
    float* dst = bufB;
    int Cin = Ctot;
    for (int l = 0; l < nl; ++l) {
        float* d = (l == nl - 1) ? out : dst;
        run_gemm(src, tot1, Cin, layers[l], Os[l], d, 1, 0, 0, stream);
        Cin = Os[l];
        src = d;
        dst = (src == bufB) ? bufA : bufB;
    }
}

extern "C" void kernel_launch(void* const* d_in, const int* in_sizes, int n_in,
                              void* d_out, int out_size, void* d_ws, size_t ws_size,
                              hipStream_t stream)
{
    (void)in_sizes; (void)n_in; (void)out_size; (void)ws_size;

    const float* xyz    = (const float*)d_in[0];   // [8,4096,3]
    const float* points = (const float*)d_in[1];   // [8,4096,3]
    Lyr sa1[3] = { getL(d_in,  2), getL(d_in,  8), getL(d_in, 14) };
    Lyr sa2[3] = { getL(d_in, 20), getL(d_in, 26), getL(d_in, 32) };
    Lyr sa3[3] = { getL(d_in, 38), getL(d_in, 44), getL(d_in, 50) };
    Lyr fp3[2] = { getL(d_in, 56), getL(d_in, 62) };
    Lyr fp2[2] = { getL(d_in, 68), getL(d_in, 74) };
    Lyr fp1[3] = { getL(d_in, 80), getL(d_in, 86), getL(d_in, 92) };
    Lyr h1     = getL(d_in, 98);
    const float* h2w = (const float*)d_in[104];    // [8,128]
    const float* h2b = (const float*)d_in[105];    // [8]

    // --- workspace bump allocator (~227 MB) ---
    char* base = (char*)d_ws;
    size_t off = 0;
    auto allocf = [&](size_t n) { float* p = (float*)(base + off); off += n * 4; return p; };
    auto alloci = [&](size_t n) { int*   p = (int*)  (base + off); off += n * 4; return p; };

    float* l1x  = allocf((size_t)8 * 1024 * 3);
    float* l2x  = allocf((size_t)8 * 256  * 3);
    float* l3x  = allocf((size_t)8 * 64   * 3);
    float* l1p  = allocf((size_t)8 * 1024 * 128);
    float* l2p  = allocf((size_t)8 * 256  * 256);
    float* l3p  = allocf((size_t)8 * 64   * 512);
    float* l0p  = allocf((size_t)8 * 4096 * 128);
    int*   fpsb = alloci((size_t)8 * 1024);
    int*   ballb= alloci((size_t)8 * 1024 * 32);
    int*   knni = alloci((size_t)8 * 4096 * 3);
    float* knnw = allocf((size_t)8 * 4096 * 3);
    float* bufA = allocf((size_t)16777216);        // 16.8M floats
    float* bufB = allocf((size_t)33554432);        // 33.6M floats

    // --- Set abstraction ---
    { int Os[3] = { 64,  64, 128};
      run_sa(xyz, 4096, points,   3, 1024, 0.1f, sa1, 3, Os, l1x, l1p, fpsb, ballb, bufA, bufB, stream); }
    { int Os[3] = {128, 128, 256};
      run_sa(l1x, 1024, l1p,    128,  256, 0.2f, sa2, 3, Os, l2x, l2p, fpsb, ballb, bufA, bufB, stream); }
    { int Os[3] = {256, 256, 512};
      run_sa(l2x,  256, l2p,    256,   64, 0.4f, sa3, 3, Os, l3x, l3p, fpsb, ballb, bufA, bufB, stream); }

    // --- Feature propagation ---
    { int Os[2] = {256, 256};
      run_fp(l2x,  256, l3x,   64, l2p, 256, l3p, 512, fp3, 2, Os, l2p, knni, knnw, bufA, bufB, stream); }
    { int Os[2] = {256, 128};
      run_fp(l1x, 1024, l2x,  256, l1p, 128, l2p, 256, fp2, 2, Os, l1p, knni, knnw, bufA, bufB, stream); }
    { int Os[3] = {128, 128, 128};
      run_fp(xyz, 4096, l1x, 1024, nullptr, 0, l1p, 128, fp1, 3, Os, l0p, knni, knnw, bufA, bufB, stream); }

    // --- Head ---
    run_gemm(l0p, 8L * 4096, 128, h1, 128, bufA, 1, 0, 0, stream);
    Lyr h2 = { h2w, h2b, h2b, h2b, h2b, h2b };     // bias-only path; BN ptrs unused
    run_gemm(bufA, 8L * 4096, 128, h2, 8, (float*)d_out, 0, 1, 4096, stream);
}